// ProbAttention_64321430225662
// MI455X (gfx1250) — compile-verified
//
#include <hip/hip_runtime.h>

#define B_  4
#define L_  2048
#define DM_ 512
#define H_  8
#define DK_ 64
#define UK_ 38
#define UQ_ 38

typedef float v2f __attribute__((ext_vector_type(2)));
typedef float v8f __attribute__((ext_vector_type(8)));
typedef unsigned int v4u __attribute__((ext_vector_type(4)));
typedef int v4i __attribute__((ext_vector_type(4)));
typedef int v8i __attribute__((ext_vector_type(8)));

#define KBLK 64   // K-slab staged in LDS per workgroup

// ---------------------------------------------------------------------------
// f32 WMMA GEMM: C(MxN) = A(MxK,row-major) * W(KxN,row-major)
// One wave -> 16x64 tile (4 accumulators); block(256)=8 waves -> 128x64 tile.
// B slab (KBLK x 64) is staged in LDS, double-buffered, filled by the Tensor
// Data Mover (tensor_load_to_lds) issued by wave 0 and synchronized with
// s_wait_tensorcnt + one workgroup barrier per slab.
// mode 0: row-major store. mode 1: scatter to (B,H,L,64) head layout.
// ---------------------------------------------------------------------------
__global__ __launch_bounds__(256) void gemm_wmma_f32(
    const float* __restrict__ A, const float* __restrict__ W,
    float* __restrict__ C, int M, int N, int K, int mode)
{
  __shared__ float Bs[2][KBLK * 64];

  const int wave = threadIdx.x >> 5;
  const int lane = threadIdx.x & 31;
  const int half = lane >> 4;   // 0: K pair {0,1}, 1: K pair {2,3}
  const int l15  = lane & 15;
  const int m0 = blockIdx.x * 128 + wave * 16;
  const int n0 = blockIdx.y * 64;

  v8f c0 = {}, c1 = {}, c2 = {}, c3 = {};
  const float* arow = A + (size_t)(m0 + l15) * K + half * 2;
  const int kblocks = K / KBLK;

#if __has_builtin(__builtin_amdgcn_tensor_load_to_lds)
  // --- TDM slab fetch: D# per CDNA5 ISA ---
  auto issue_tdm = [&](int buf, int kb) {
    unsigned long long gaddr =
        (unsigned long long)(uintptr_t)(W + (size_t)(kb * KBLK) * N + n0);
    __attribute__((address_space(3))) float* p3 =
        (__attribute__((address_space(3))) float*)&Bs[buf][0];
    unsigned lds_addr = (unsigned)(unsigned long long)p3;
    v4u g0;
    g0.x = 1u;                                   // count=1, user descriptor
    g0.y = lds_addr;                             // lds_addr (bytes)
    g0.z = (unsigned)(gaddr & 0xFFFFFFFFu);      // global_addr[31:0]
    g0.w = (unsigned)((gaddr >> 32) & 0x01FFFFFFu) | (2u << 30); // addr[56:32] | type=2
    v8i g1;
    g1[0] = 2 << 16;                                   // data_size=4B, no mask/pad
    g1[1] = (int)(((unsigned)N & 0xFFFFu) << 16);      // tensor_dim0[15:0]
    g1[2] = (int)((((unsigned)N >> 16) & 0xFFFFu) |    // tensor_dim0[31:16]
                  (((unsigned)K & 0xFFFFu) << 16));    // tensor_dim1[15:0]
    g1[3] = (int)((((unsigned)K >> 16) & 0xFFFFu) |    // tensor_dim1[31:16]
                  ((unsigned)64 << 16));               // tile_dim0 = 64
    g1[4] = KBLK;                                      // tile_dim1 = KBLK (tile_dim2=0)
    g1[5] = N;                                         // tensor_dim0_stride[31:0]
    g1[6] = 0;
    g1[7] = 0;
    v4i z4 = {0, 0, 0, 0};
    v8i z8 = {0, 0, 0, 0, 0, 0, 0, 0};
    __builtin_amdgcn_tensor_load_to_lds(g0, g1, z4, z4, z8, 0);
  };

  if (wave == 0) issue_tdm(0, 0);
  for (int kb = 0; kb < kblocks; ++kb) {
    if (wave == 0) __builtin_amdgcn_s_wait_tensorcnt(0); // slab kb landed
    __syncthreads();                                      // visible to all; prev compute done
    if (wave == 0 && kb + 1 < kblocks) issue_tdm((kb + 1) & 1, kb + 1);
    const float* Bc = Bs[kb & 1];
#else
  for (int kb = 0; kb < kblocks; ++kb) {
    __syncthreads();
    {  // cooperative synchronous slab fill (fallback path)
      const float* src = W + (size_t)(kb * KBLK) * N + n0;
      for (int i = threadIdx.x; i < KBLK * 16; i += 256) {
        const int r = i >> 4, c4 = (i & 15) << 2;
        ((float4*)&Bs[kb & 1][r * 64 + c4])[0] =
            *(const float4*)(src + (size_t)r * N + c4);
      }
    }
    __syncthreads();
    const float* Bc = Bs[kb & 1];
#endif

    #pragma unroll 4
    for (int kk = 0; kk < KBLK; kk += 4) {
      const int k0 = kb * KBLK + kk;
      v2f a;
      a.x = arow[k0];
      a.y = arow[k0 + 1];
      const float* b0p = Bc + (kk + half * 2) * 64 + l15;
      const float* b1p = b0p + 64;
      v2f b0, b1, b2, b3;
      b0.x = b0p[0];  b0.y = b1p[0];
      b1.x = b0p[16]; b1.y = b1p[16];
      b2.x = b0p[32]; b2.y = b1p[32];
      b3.x = b0p[48]; b3.y = b1p[48];
      c0 = __builtin_amdgcn_wmma_f32_16x16x4_f32(false, a, false, b0, (short)0, c0, false, false);
      c1 = __builtin_amdgcn_wmma_f32_16x16x4_f32(false, a, false, b1, (short)0, c1, false, false);
      c2 = __builtin_amdgcn_wmma_f32_16x16x4_f32(false, a, false, b2, (short)0, c2, false, false);
      c3 = __builtin_amdgcn_wmma_f32_16x16x4_f32(false, a, false, b3, (short)0, c3, false, false);
    }
  }

  #pragma unroll
  for (int i = 0; i < 8; ++i) {
    const int row = m0 + i + half * 8;
    const int col = n0 + l15;
    float v[4] = { c0[i], c1[i], c2[i], c3[i] };
    #pragma unroll
    for (int t = 0; t < 4; ++t) {
      const int cc = col + t * 16;
      if (mode == 0) {
        C[(size_t)row * N + cc] = v[t];
      } else {
        const int b = row / L_, l = row % L_;
        const int h = cc >> 6, d = cc & 63;
        C[(((size_t)(b * H_ + h)) * L_ + l) * DK_ + d] = v[t];
      }
    }
  }
}

// ---------------------------------------------------------------------------
// M[b,h,q] = max_k(Q.K_sample) - mean_k(Q.K_sample), one wave per (b,h,q)
// ---------------------------------------------------------------------------
__global__ __launch_bounds__(256) void qk_sample_M(
    const float* __restrict__ Q, const float* __restrict__ K,
    const int* __restrict__ samp, float* __restrict__ Mout)
{
  const int wid  = blockIdx.x * 8 + (threadIdx.x >> 5);
  const int lane = threadIdx.x & 31;
  const int bh = wid / L_;
  const int q  = wid % L_;
  const float* qp = Q + ((size_t)bh * L_ + q) * DK_;
  const float q0 = qp[lane], q1 = qp[lane + 32];
  float mx = -3.4e38f, sm = 0.f;
  for (int k = 0; k < UK_; ++k) {
    const int j = samp[q * UK_ + k];
    const float* kp = K + ((size_t)bh * L_ + j) * DK_;
    float p = q0 * kp[lane] + q1 * kp[lane + 32];
    #pragma unroll
    for (int off = 16; off; off >>= 1) p += __shfl_xor(p, off, 32);
    mx = fmaxf(mx, p);
    sm += p;
  }
  if (lane == 0) Mout[wid] = mx - sm * (1.0f / UK_);
}

// ---------------------------------------------------------------------------
// top-38 per (b,h): iterative arg-max, ties -> lower index (JAX semantics)
// ---------------------------------------------------------------------------
__global__ __launch_bounds__(256) void topk38(
    const float* __restrict__ Mbuf, int* __restrict__ idx_ws,
    int* __restrict__ idx_out)
{
  __shared__ float vals[L_];
  __shared__ float rv[256];
  __shared__ int   ri[256];
  const int bh  = blockIdx.x;
  const int tid = threadIdx.x;
  for (int i = tid; i < L_; i += 256) vals[i] = Mbuf[(size_t)bh * L_ + i];
  __syncthreads();
  for (int s = 0; s < UQ_; ++s) {
    float bv = -3.4e38f; int bi = L_;
    for (int i = tid; i < L_; i += 256) {
      const float v = vals[i];
      if (v > bv || (v == bv && i < bi)) { bv = v; bi = i; }
    }
    rv[tid] = bv; ri[tid] = bi;
    __syncthreads();
    for (int st = 128; st; st >>= 1) {
      if (tid < st) {
        const float ov = rv[tid + st]; const int oi = ri[tid + st];
        if (ov > rv[tid] || (ov == rv[tid] && oi < ri[tid])) { rv[tid] = ov; ri[tid] = oi; }
      }
      __syncthreads();
    }
    if (tid == 0) {
      const int w = ri[0];
      idx_ws[bh * UQ_ + s]  = w;
      idx_out[bh * UQ_ + s] = w;
      vals[w] = -3.4e38f;
    }
    __syncthreads();
  }
}

// ---------------------------------------------------------------------------
// softmax(Q_sel.K^T / 8) @ V  -> values[b,h,u,64], one block per (b,h,u)
// ---------------------------------------------------------------------------
__global__ __launch_bounds__(256) void attn_kernel(
    const float* __restrict__ Q, const float* __restrict__ Kb,
    const float* __restrict__ Vb, const int* __restrict__ idx_ws,
    float* __restrict__ values)
{
  __shared__ float qv[DK_];
  __shared__ float sc[L_];
  __shared__ float red[256];
  const int bh  = blockIdx.x / UQ_;
  const int u   = blockIdx.x % UQ_;
  const int tid = threadIdx.x;
  const int qidx = idx_ws[bh * UQ_ + u];
  if (tid < DK_) qv[tid] = Q[((size_t)bh * L_ + qidx) * DK_ + tid];
  __syncthreads();

  float lmax = -3.4e38f;
  for (int k = tid; k < L_; k += 256) {
    const float* kp = Kb + ((size_t)bh * L_ + k) * DK_;
    float acc = 0.f;
    #pragma unroll
    for (int d = 0; d < DK_; ++d) acc += qv[d] * kp[d];
    acc *= 0.125f;              // 1/sqrt(64)
    sc[k] = acc;
    lmax = fmaxf(lmax, acc);
  }
  red[tid] = lmax; __syncthreads();
  for (int st = 128; st; st >>= 1) {
    if (tid < st) red[tid] = fmaxf(red[tid], red[tid + st]);
    __syncthreads();
  }
  const float gmax = red[0];
  __syncthreads();

  float lsum = 0.f;
  for (int k = tid; k < L_; k += 256) {
    const float e = __expf(sc[k] - gmax);
    sc[k] = e;
    lsum += e;
  }
  red[tid] = lsum; __syncthreads();
  for (int st = 128; st; st >>= 1) {
    if (tid < st) red[tid] += red[tid + st];
    __syncthreads();
  }
  const float inv = 1.0f / red[0];
  __syncthreads();

  const int d = tid & 63, c = tid >> 6;       // 4 chunks of 512 keys
  float acc = 0.f;
  for (int k = c * 512; k < (c + 1) * 512; ++k)
    acc += sc[k] * Vb[((size_t)bh * L_ + k) * DK_ + d];
  red[tid] = acc; __syncthreads();
  if (tid < 64) {
    const float s = red[tid] + red[tid + 64] + red[tid + 128] + red[tid + 192];
    values[((size_t)bh * UQ_ + u) * DK_ + tid] = s * inv;
  }
}

// ---------------------------------------------------------------------------
// vmean[b,h,d] = mean over L of V
// ---------------------------------------------------------------------------
__global__ __launch_bounds__(256) void vmean_kernel(
    const float* __restrict__ Vb, float* __restrict__ vmean)
{
  __shared__ float red[256];
  const int bh = blockIdx.x;
  const int tid = threadIdx.x;
  const int d = tid & 63, c = tid >> 6;
  float acc = 0.f;
  for (int k = c * 512; k < (c + 1) * 512; ++k)
    acc += Vb[((size_t)bh * L_ + k) * DK_ + d];
  red[tid] = acc; __syncthreads();
  if (tid < 64)
    vmean[bh * DK_ + tid] =
        (red[tid] + red[tid + 64] + red[tid + 128] + red[tid + 192]) * (1.0f / L_);
}

// ctx[b,l,h*64+d] = vmean[b,h,d]
__global__ __launch_bounds__(256) void ctx_fill(
    const float* __restrict__ vmean, float* __restrict__ ctx)
{
  const size_t e = (size_t)blockIdx.x * 256 + threadIdx.x;   // < B*L*512
  const int c = (int)(e & 511);
  const size_t bl = e >> 9;
  const int b = (int)(bl >> 11);                             // bl / L_
  const int h = c >> 6, d = c & 63;
  ctx[e] = vmean[(b * H_ + h) * DK_ + d];
}

// ctx[b, idx, h*64+d] = values[b,h,u,d]
__global__ __launch_bounds__(64) void ctx_scatter(
    const float* __restrict__ values, const int* __restrict__ idx_ws,
    float* __restrict__ ctx)
{
  const int bh = blockIdx.x / UQ_, u = blockIdx.x % UQ_;
  const int b = bh / H_, h = bh % H_;
  const int d = threadIdx.x;
  const int l = idx_ws[bh * UQ_ + u];
  ctx[((size_t)b * L_ + l) * DM_ + h * DK_ + d] =
      values[((size_t)bh * UQ_ + u) * DK_ + d];
}

// ---------------------------------------------------------------------------
// y = LayerNorm(fc + input_Q) * gamma + beta, one block per row of 512
// ---------------------------------------------------------------------------
__global__ __launch_bounds__(256) void resid_ln(
    const float* __restrict__ fc, const float* __restrict__ inq,
    const float* __restrict__ gamma, const float* __restrict__ beta,
    float* __restrict__ out)
{
  __shared__ float rs[256], rq[256];
  const size_t row = blockIdx.x;
  const int tid = threadIdx.x;
  const float y0 = fc[row * DM_ + tid]       + inq[row * DM_ + tid];
  const float y1 = fc[row * DM_ + tid + 256] + inq[row * DM_ + tid + 256];
  rs[tid] = y0 + y1;
  rq[tid] = y0 * y0 + y1 * y1;
  __syncthreads();
  for (int st = 128; st; st >>= 1) {
    if (tid < st) { rs[tid] += rs[tid + st]; rq[tid] += rq[tid + st]; }
    __syncthreads();
  }
  const float mu  = rs[0] * (1.0f / DM_);
  const float var = rq[0] * (1.0f / DM_) - mu * mu;
  const float inv = rsqrtf(var + 1e-5f);
  out[row * DM_ + tid]       = (y0 - mu) * inv * gamma[tid]       + beta[tid];
  out[row * DM_ + tid + 256] = (y1 - mu) * inv * gamma[tid + 256] + beta[tid + 256];
}

// ---------------------------------------------------------------------------
extern "C" void kernel_launch(void* const* d_in, const int* in_sizes, int n_in,
                              void* d_out, int out_size, void* d_ws, size_t ws_size,
                              hipStream_t stream)
{
  const float* inQ   = (const float*)d_in[0];
  const float* inK   = (const float*)d_in[1];
  const float* inV   = (const float*)d_in[2];
  const float* WQ    = (const float*)d_in[3];
  const float* WK    = (const float*)d_in[4];
  const float* WV    = (const float*)d_in[5];
  const float* Wfc   = (const float*)d_in[6];
  const float* gamma = (const float*)d_in[7];
  const float* beta  = (const float*)d_in[8];
  const int*   samp  = (const int*)d_in[9];

  float* out = (float*)d_out;
  int* idx_out = (int*)(out + (size_t)B_ * L_ * DM_);   // tuple elem 2 (int32)

  char* ws = (char*)d_ws;
  size_t off = 0;
  auto alloc = [&](size_t bytes) -> char* {
    char* p = ws + off;
    off += (bytes + 255) & ~(size_t)255;
    return p;
  };
  float* Qb     = (float*)alloc((size_t)B_ * H_ * L_ * DK_ * 4);
  float* Kb     = (float*)alloc((size_t)B_ * H_ * L_ * DK_ * 4);
  float* Vb     = (float*)alloc((size_t)B_ * H_ * L_ * DK_ * 4);
  float* Mb     = (float*)alloc((size_t)B_ * H_ * L_ * 4);
  int*   idx_ws = (int*)  alloc((size_t)B_ * H_ * UQ_ * 4);
  float* values = (float*)alloc((size_t)B_ * H_ * UQ_ * DK_ * 4);
  float* vmean  = (float*)alloc((size_t)B_ * H_ * DK_ * 4);
  float* ctx    = (float*)alloc((size_t)B_ * L_ * DM_ * 4);
  float* fco    = (float*)alloc((size_t)B_ * L_ * DM_ * 4);

  const dim3 gg(B_ * L_ / 128, DM_ / 64);
  gemm_wmma_f32<<<gg, 256, 0, stream>>>(inQ, WQ, Qb, B_ * L_, DM_, DM_, 1);
  gemm_wmma_f32<<<gg, 256, 0, stream>>>(inK, WK, Kb, B_ * L_, DM_, DM_, 1);
  gemm_wmma_f32<<<gg, 256, 0, stream>>>(inV, WV, Vb, B_ * L_, DM_, DM_, 1);

  qk_sample_M<<<B_ * H_ * L_ / 8, 256, 0, stream>>>(Qb, Kb, samp, Mb);
  topk38<<<B_ * H_, 256, 0, stream>>>(Mb, idx_ws, idx_out);
  attn_kernel<<<B_ * H_ * UQ_, 256, 0, stream>>>(Qb, Kb, Vb, idx_ws, values);
  vmean_kernel<<<B_ * H_, 256, 0, stream>>>(Vb, vmean);
  ctx_fill<<<(B_ * L_ * DM_) / 256, 256, 0, stream>>>(vmean, ctx);
  ctx_scatter<<<B_ * H_ * UQ_, 64, 0, stream>>>(values, idx_ws, ctx);

  gemm_wmma_f32<<<gg, 256, 0, stream>>>(ctx, Wfc, fco, B_ * L_, DM_, DM_, 0);
  resid_ln<<<B_ * L_, 256, 0, stream>>>(fco, inQ, gamma, beta, out);
}